// GNN_75239237091885
// MI455X (gfx1250) — compile-verified
//
#include <hip/hip_runtime.h>
#include <hip/hip_bf16.h>

// ---------------------------------------------------------------------------
// GNN: 3x GraphConv(64->64) + global mean pool + linear, fp32 end to end.
// Aggregation = L2-resident atomic scatter; dense parts = v_wmma_f32_16x16x4_f32
// with A-tiles AND weights staged in LDS (no global traffic in the hot loop).
// ---------------------------------------------------------------------------

typedef float v2f __attribute__((ext_vector_type(2)));
typedef float v8f __attribute__((ext_vector_type(8)));

#define D 64                    // feature width (D_IN == D_HID == 64)
#define A_STRIDE 132            // 128 K-columns + 4 pad floats (bank-conflict free)
#define W_STRIDE 66             // 64 K-columns + 2 pad floats (<=2-way conflicts)
#define W_MAT    (64 * W_STRIDE)          // floats per staged weight matrix
#define W_FLOATS (2 * W_MAT)              // Wr + Wn
#define WAVES_PER_BLOCK 8
#define ROWS_PER_BLOCK (WAVES_PER_BLOCK * 16)
#define A_FLOATS (WAVES_PER_BLOCK * 16 * A_STRIDE)
#define CONV_SMEM_BYTES ((W_FLOATS + A_FLOATS) * 4)

// ---------------------------------------------------------------------------
// Edge scatter: aggr[dst] += h[src].  One wave32 per edge, 2 features/lane.
// Gather is one coalesced 256B row read per wave; adds hit the 25.6MB
// node accumulator which lives entirely in the 192MB L2.
// ---------------------------------------------------------------------------
__global__ __launch_bounds__(256) void gnn_scatter_kernel(
    const float* __restrict__ h,
    const int*   __restrict__ src,
    const int*   __restrict__ dst,
    float*       __restrict__ aggr,
    int n_edges)
{
    const int e    = blockIdx.x * (blockDim.x >> 5) + (threadIdx.x >> 5);
    const int lane = threadIdx.x & 31;
    if (e >= n_edges) return;
    const int s = src[e];
    const int d = dst[e];
    const float2 v = *(const float2*)&h[(size_t)s * D + lane * 2];
    atomicAdd(&aggr[(size_t)d * D + lane * 2 + 0], v.x);
    atomicAdd(&aggr[(size_t)d * D + lane * 2 + 1], v.y);
}

// ---------------------------------------------------------------------------
// Fused GraphConv GEMM: out = aggr @ Wr^T + b + h @ Wn^T  (optional ReLU)
// Each wave computes a 16x64 output tile via 4 x (32 x wmma_f32_16x16x4_f32).
// A-tile (16 x [aggr|h] = 16x128) per wave in LDS (conflict-free float2 frags),
// both weight matrices staged in LDS once per block (coalesced), B-fragments
// read as ds_load_b64.
// ---------------------------------------------------------------------------
__global__ __launch_bounds__(256) void gnn_conv_kernel(
    const float* __restrict__ aggr,   // [N][64]
    const float* __restrict__ h,      // [N][64]
    const float* __restrict__ Wr,     // [64][64] row-major (out x in)
    const float* __restrict__ Wn,     // [64][64]
    const float* __restrict__ bias,   // [64]
    float*       __restrict__ out,    // [N][64]
    int n_nodes, int do_relu)
{
    extern __shared__ float smem[];   // [weights | 8 waves x 16 x A_STRIDE]
    float* ws    = smem;              // [2][64][W_STRIDE]
    float* tiles = smem + W_FLOATS;

    const int wave = threadIdx.x >> 5;
    const int lane = threadIdx.x & 31;
    const int half = lane >> 4;       // 0: K+0/1, M rows 0..7 ; 1: K+2/3, M rows 8..15
    const int nl   = lane & 15;
    const int row0 = (blockIdx.x * WAVES_PER_BLOCK + wave) * 16;

    float* tile = tiles + wave * 16 * A_STRIDE;

    // ---- Stage both weight matrices (coalesced global reads, padded LDS rows)
    for (int i = threadIdx.x; i < 64 * 64; i += 256) {
        const int n = i >> 6, k = i & 63;
        ws[n * W_STRIDE + k]         = Wr[i];
        ws[W_MAT + n * W_STRIDE + k] = Wn[i];
    }

    // ---- Stage A-tile: lanes 0-15 load aggr cols (float4), lanes 16-31 load h.
    for (int r = 0; r < 16; ++r) {
        const int row = row0 + r;
        float4 v = make_float4(0.f, 0.f, 0.f, 0.f);
        if (row < n_nodes) {
            const float* srcp = (half == 0) ? &aggr[(size_t)row * D + nl * 4]
                                            : &h[(size_t)row * D + nl * 4];
            v = *(const float4*)srcp;
        }
        *(float4*)&tile[r * A_STRIDE + half * 64 + nl * 4] = v;
    }
    __syncthreads();

    // ---- 4 output column tiles of 16
    for (int nt = 0; nt < 4; ++nt) {
        const int n = nt * 16 + nl;
        const float bv = bias[n];
        v8f c;
        #pragma unroll
        for (int i = 0; i < 8; ++i) c[i] = bv;

        // K = 0..127 in steps of 4 (K<64 -> aggr x Wr ; K>=64 -> h x Wn)
        #pragma unroll
        for (int kk = 0; kk < 32; ++kk) {
            const int kb = kk * 4;
            // A fragment: lane half selects K+0/1 vs K+2/3 of row M=nl
            v2f a = *(const v2f*)&tile[nl * A_STRIDE + kb + 2 * half];
            // B fragment: B[k][n] = W[n][k], from LDS-staged weights
            const float* Wb = ws + ((kb < 64) ? 0 : W_MAT);
            const int    kw = kb & 63;
            v2f b = *(const v2f*)&Wb[n * W_STRIDE + kw + 2 * half];
            c = __builtin_amdgcn_wmma_f32_16x16x4_f32(
                    false, a, false, b, (short)0, c, false, false);
        }

        // D layout: VGPR i -> M = i + 8*half, N = nl (+16*nt)
        #pragma unroll
        for (int i = 0; i < 8; ++i) {
            const int row = row0 + i + 8 * half;
            if (row < n_nodes) {
                float v = c[i];
                if (do_relu) v = fmaxf(v, 0.f);
                out[(size_t)row * D + n] = v;
            }
        }
    }
}

// ---------------------------------------------------------------------------
// Mean-pool accumulation: pooled[batch[node]] += h[node]; counts[g] += 1
// ---------------------------------------------------------------------------
__global__ __launch_bounds__(256) void gnn_pool_kernel(
    const float* __restrict__ h,
    const int*   __restrict__ batch,
    float*       __restrict__ pooled,   // [64][64]
    float*       __restrict__ counts,   // [64]
    int n_nodes)
{
    const int node = blockIdx.x * (blockDim.x >> 5) + (threadIdx.x >> 5);
    const int lane = threadIdx.x & 31;
    if (node >= n_nodes) return;
    const int g = batch[node];
    const float2 v = *(const float2*)&h[(size_t)node * D + lane * 2];
    atomicAdd(&pooled[g * D + lane * 2 + 0], v.x);
    atomicAdd(&pooled[g * D + lane * 2 + 1], v.y);
    if (lane == 0) atomicAdd(&counts[g], 1.0f);
}

// ---------------------------------------------------------------------------
// Final linear on pooled means: out[g][n] = bl[n] + sum_k mean[g][k]*Wl[n][k]
// 64x64 output, 0.5 MFLOP -> scalar is fine.
// ---------------------------------------------------------------------------
__global__ __launch_bounds__(64) void gnn_final_kernel(
    const float* __restrict__ pooled,
    const float* __restrict__ counts,
    const float* __restrict__ Wl,
    const float* __restrict__ bl,
    float*       __restrict__ out)
{
    const int g = blockIdx.x;    // 64 graphs
    const int n = threadIdx.x;   // 64 outputs
    const float inv = 1.0f / fmaxf(counts[g], 1.0f);
    float acc = bl[n];
    #pragma unroll
    for (int k = 0; k < D; ++k)
        acc += pooled[g * D + k] * inv * Wl[n * D + k];
    out[g * D + n] = acc;
}

// ---------------------------------------------------------------------------
extern "C" void kernel_launch(void* const* d_in, const int* in_sizes, int n_in,
                              void* d_out, int out_size, void* d_ws, size_t ws_size,
                              hipStream_t stream)
{
    const float* x     = (const float*)d_in[0];
    const int*   ei    = (const int*)d_in[1];
    const int*   batch = (const int*)d_in[2];
    const float* W1r = (const float*)d_in[3];
    const float* b1  = (const float*)d_in[4];
    const float* W1n = (const float*)d_in[5];
    const float* W2r = (const float*)d_in[6];
    const float* b2  = (const float*)d_in[7];
    const float* W2n = (const float*)d_in[8];
    const float* W3r = (const float*)d_in[9];
    const float* b3  = (const float*)d_in[10];
    const float* W3n = (const float*)d_in[11];
    const float* Wl  = (const float*)d_in[12];
    const float* bl  = (const float*)d_in[13];

    const int n_nodes = in_sizes[0] / D;
    const int n_edges = in_sizes[1] / 2;
    const int*   src  = ei;
    const int*   dst  = ei + n_edges;

    // Workspace layout
    const size_t node_f = (size_t)n_nodes * D;           // floats per node buffer
    float* aggr   = (float*)d_ws;
    float* h1     = aggr + node_f;
    float* h2     = h1 + node_f;
    float* h3     = h2 + node_f;
    float* pooled = h3 + node_f;                         // 64*64 floats
    float* counts = pooled + 64 * D;                     // 64 floats
    (void)ws_size; (void)n_in; (void)out_size;

    const dim3 blk256(256);
    const int scatter_grid = (n_edges + 7) / 8;                         // 8 edges/block
    const int conv_grid    = (n_nodes + ROWS_PER_BLOCK - 1) / ROWS_PER_BLOCK;
    const int pool_grid    = (n_nodes + 7) / 8;                         // 8 nodes/block

    // ---- layer 1: relu(aggr(x) @ W1r^T + b1 + x @ W1n^T) ----
    hipMemsetAsync(aggr, 0, node_f * sizeof(float), stream);
    gnn_scatter_kernel<<<scatter_grid, blk256, 0, stream>>>(x, src, dst, aggr, n_edges);
    gnn_conv_kernel<<<conv_grid, blk256, CONV_SMEM_BYTES, stream>>>(
        aggr, x, W1r, W1n, b1, h1, n_nodes, 1);

    // ---- layer 2 ----
    hipMemsetAsync(aggr, 0, node_f * sizeof(float), stream);
    gnn_scatter_kernel<<<scatter_grid, blk256, 0, stream>>>(h1, src, dst, aggr, n_edges);
    gnn_conv_kernel<<<conv_grid, blk256, CONV_SMEM_BYTES, stream>>>(
        aggr, h1, W2r, W2n, b2, h2, n_nodes, 1);

    // ---- layer 3 (no relu) ----
    hipMemsetAsync(aggr, 0, node_f * sizeof(float), stream);
    gnn_scatter_kernel<<<scatter_grid, blk256, 0, stream>>>(h2, src, dst, aggr, n_edges);
    gnn_conv_kernel<<<conv_grid, blk256, CONV_SMEM_BYTES, stream>>>(
        aggr, h2, W3r, W3n, b3, h3, n_nodes, 0);

    // ---- mean pool + final linear ----
    hipMemsetAsync(pooled, 0, (64 * D + 64) * sizeof(float), stream);
    gnn_pool_kernel<<<pool_grid, blk256, 0, stream>>>(h3, batch, pooled, counts, n_nodes);
    gnn_final_kernel<<<64, 64, 0, stream>>>(pooled, counts, Wl, bl, (float*)d_out);
}